// HeatmapEncoder_22247930593771
// MI455X (gfx1250) — compile-verified
//
#include <hip/hip_runtime.h>

typedef __attribute__((ext_vector_type(2))) float v2f;
typedef __attribute__((ext_vector_type(8))) float v8f;

#define HM_H 336
#define HM_W 336
#define HM_TILES_X 21              // 336/16
#define HM_TILES (21 * 21)         // 441 tiles of 16x16 per slice
#define NTHREADS 256
#define NWAVES (NTHREADS / 32)

__global__ __launch_bounds__(NTHREADS)
void HeatmapEncoder_kernel(const float* __restrict__ gaze,
                           const float* __restrict__ hand,
                           float* __restrict__ out) {
    constexpr float SIGMA  = 10.0f / 336.0f;
    constexpr float INV2S2 = 1.0f / (2.0f * SIGMA * SIGMA);
    constexpr float EPS    = 1e-6f;
    constexpr float INVWM1 = 1.0f / (HM_W - 1);

    __shared__ float fx1[HM_W], fx2[HM_W], fy1[HM_H], fy2[HM_H];
    __shared__ float sums[4];            // Σfx1, Σfx2, Σfy1, Σfy2
    __shared__ unsigned int peakBits;

    const int tid  = threadIdx.x;
    const int lane = tid & 31;
    const int bsc  = blockIdx.x;         // (b,s,c) flat index, 0..255
    // Scalarize the wave index so tile loops are SALU-controlled.
    const int waveS = __builtin_amdgcn_readfirstlane(tid >> 5);

    if (tid < 4) sums[tid] = 0.0f;
    if (tid == 4) peakBits = 0u;

    // Per-slice coordinates (x, y), normalized
    const float cx1 = gaze[bsc * 2 + 0];
    const float cy1 = gaze[bsc * 2 + 1];
    const float cx2 = hand[bsc * 2 + 0];
    const float cy2 = hand[bsc * 2 + 1];
    const bool valid1 = (cx1 + cy1) > 0.0f;
    const bool valid2 = (cx2 + cy2) > 0.0f;

    // ---- Phase 1: separable Gaussian factors into LDS + partial sums ----
    float s0 = 0.0f, s1 = 0.0f, s2 = 0.0f, s3 = 0.0f;
    for (int i = tid; i < HM_W; i += NTHREADS) {
        float g  = (float)i * INVWM1;
        float d;
        d = g - cx1; float e0 = __expf(-(d * d) * INV2S2);
        d = g - cx2; float e1 = __expf(-(d * d) * INV2S2);
        d = g - cy1; float e2 = __expf(-(d * d) * INV2S2);
        d = g - cy2; float e3 = __expf(-(d * d) * INV2S2);
        fx1[i] = e0; fx2[i] = e1; fy1[i] = e2; fy2[i] = e3;
        s0 += e0; s1 += e1; s2 += e2; s3 += e3;
    }
    #pragma unroll
    for (int off = 16; off > 0; off >>= 1) {
        s0 += __shfl_xor(s0, off, 32);
        s1 += __shfl_xor(s1, off, 32);
        s2 += __shfl_xor(s2, off, 32);
        s3 += __shfl_xor(s3, off, 32);
    }
    if (lane == 0) {
        atomicAdd(&sums[0], s0);
        atomicAdd(&sums[1], s1);
        atomicAdd(&sums[2], s2);
        atomicAdd(&sums[3], s3);
    }
    __syncthreads();

    // 2D sum of each separable Gaussian = Σfx * Σfy
    const float S1 = sums[0] * sums[2];
    const float S2 = sums[1] * sums[3];
    const float a1 = valid1 ? 1.0f / (S1 + EPS) : 0.0f;
    const float a2 = valid2 ? 1.0f / (S2 + EPS) : 0.0f;

    // ---- WMMA rank-2 outer-product tile: D = A(16x4) x B(4x16) ----
    // A layout (32-bit 16x4): lanes 0-15 -> M=lane, VGPR0/1 = K0/K1; lanes 16-31 = K2/K3
    // B layout (32-bit 4x16): lanes 0-15 -> N=lane, VGPR0/1 = K0/K1; lanes 16-31 = K2/K3
    // K=2,3 are zeroed via multiplicative lane mask (no EXEC-divergent branches;
    // all LDS addresses are valid for every lane).
    const int   nIdx = lane & 15;
    const float m    = (lane < 16) ? 1.0f : 0.0f;

    auto tile = [&](int y0, int x0, float aA1, float aA2) -> v8f {
        v2f a, b;
        a.x = aA1 * fy1[y0 + nIdx];
        a.y = aA2 * fy2[y0 + nIdx];
        b.x = m * fx1[x0 + nIdx];
        b.y = m * fx2[x0 + nIdx];
        v8f c = {};
        return __builtin_amdgcn_wmma_f32_16x16x4_f32(
            false, a, false, b, (short)0, c, false, false);
    };

    // ---- Phase 2: peak over the unified heatmap (WMMA tiles, max-reduce) ----
    {
        const float aA1 = m * a1, aA2 = m * a2;
        float lmax = 0.0f;
        int tx = waveS, ty = 0;
        for (int t = waveS; t < HM_TILES; t += NWAVES) {
            v8f d = tile(ty * 16, tx * 16, aA1, aA2);
            #pragma unroll
            for (int r = 0; r < 8; ++r) lmax = fmaxf(lmax, d[r]);
            tx += NWAVES;
            if (tx >= HM_TILES_X) { tx -= HM_TILES_X; ++ty; }
        }
        #pragma unroll
        for (int off = 16; off > 0; off >>= 1)
            lmax = fmaxf(lmax, __shfl_xor(lmax, off, 32));
        if (lane == 0) atomicMax(&peakBits, __float_as_uint(lmax)); // values >= 0
    }
    __syncthreads();

    const float peak = __uint_as_float(peakBits);
    const float invP = 1.0f / (peak + EPS);

    // ---- Phase 3: reconstruct tiles with peak folded in, NT-stream to HBM ----
    // D layout: VGPR r, lane l -> row = 16*ty + r + 8*(l>=16), col = 16*tx + (l&15)
    {
        const float aA1 = m * a1 * invP, aA2 = m * a2 * invP;
        float* __restrict__ outB = out + (size_t)bsc * (HM_H * HM_W);
        const int rowOff = (lane >> 4) << 3;   // 0 or 8
        int tx = waveS, ty = 0;
        for (int t = waveS; t < HM_TILES; t += NWAVES) {
            v8f d = tile(ty * 16, tx * 16, aA1, aA2);
            // per-lane base for this tile; the 8 row stores become imm offsets
            float* __restrict__ p =
                outB + (size_t)(ty * 16 + rowOff) * HM_W + (tx * 16 + nIdx);
            #pragma unroll
            for (int r = 0; r < 8; ++r)
                __builtin_nontemporal_store(d[r], p + r * HM_W);
            tx += NWAVES;
            if (tx >= HM_TILES_X) { tx -= HM_TILES_X; ++ty; }
        }
    }
}

extern "C" void kernel_launch(void* const* d_in, const int* in_sizes, int n_in,
                              void* d_out, int out_size, void* d_ws, size_t ws_size,
                              hipStream_t stream) {
    const float* gaze = (const float*)d_in[0];   // [8,8,4,2] f32
    const float* hand = (const float*)d_in[1];   // [8,8,4,2] f32
    float* out = (float*)d_out;                  // [8,8,4,336,336] f32

    const int nSlices = 8 * 8 * 4;               // 256 workgroups, one per (b,s,c)
    HeatmapEncoder_kernel<<<nSlices, NTHREADS, 0, stream>>>(gaze, hand, out);
    (void)in_sizes; (void)n_in; (void)out_size; (void)d_ws; (void)ws_size;
}